// FraudGNN_84439057039711
// MI455X (gfx1250) — compile-verified
//
#include <hip/hip_runtime.h>
#include <hip/hip_bf16.h>

typedef __attribute__((ext_vector_type(2))) float v2f;
typedef __attribute__((ext_vector_type(8))) float v8f;
typedef __attribute__((ext_vector_type(4))) unsigned int v4u;
typedef __attribute__((ext_vector_type(8))) int v8i;
typedef __attribute__((ext_vector_type(4))) int v4i;

#define IN_CH 8
#define HID   64
#define EMB   32

// ---------------------------------------------------------------------------
// Workspace zeroing (aggr1 | cnt | aggr2 region)
// ---------------------------------------------------------------------------
__global__ void k_zero(float* __restrict__ p, long long n) {
    long long i = (long long)blockIdx.x * blockDim.x + threadIdx.x;
    if (i < n) p[i] = 0.0f;
}

// ---------------------------------------------------------------------------
// Layer-1 edge scatter: aggr1[dst] += x[src] (8 ch), cnt[dst] += 1
// ---------------------------------------------------------------------------
__global__ void k_edge1(const float* __restrict__ x,
                        const int* __restrict__ eidx, int E,
                        float* __restrict__ aggr1, float* __restrict__ cnt) {
    long long tid = (long long)blockIdx.x * blockDim.x + threadIdx.x;
    if (tid >= (long long)E * 2) return;
    int e = (int)(tid >> 1);
    int g = (int)(tid & 1);
    int src = eidx[e];
    int dst = eidx[E + e];
    float4 v = reinterpret_cast<const float4*>(x)[(long long)src * 2 + g];
    float* a = aggr1 + (long long)dst * IN_CH + g * 4;
    atomicAdd(a + 0, v.x);
    atomicAdd(a + 1, v.y);
    atomicAdd(a + 2, v.z);
    atomicAdd(a + 3, v.w);
    if (g == 0) atomicAdd(&cnt[dst], 1.0f);
}

// ---------------------------------------------------------------------------
// GEMM1: h = relu([aggr1/cnt | x] (Nx16) @ [W1l;W1r] (16x64) + b1)
// One wave per 16-row tile, K=16 -> 4 WMMA f32 16x16x4 per col-tile.
// ---------------------------------------------------------------------------
__global__ void k_gemm1(const float* __restrict__ x,
                        const float* __restrict__ aggr1,
                        const float* __restrict__ cnt,
                        const float* __restrict__ W1l,   // [8,64]
                        const float* __restrict__ b1,    // [64]
                        const float* __restrict__ W1r,   // [8,64]
                        float* __restrict__ h, int N) {
    const int lane = threadIdx.x & 31;
    const int wave = threadIdx.x >> 5;
    const int m  = lane & 15;     // row within tile (A) / col within tile (B)
    const int kh = lane >> 4;     // k-half select

    // B fragments + bias, loaded once and reused across tiles
    v2f bf[4][4];
    float bias[4];
#pragma unroll
    for (int kt = 0; kt < 4; ++kt) {
        int k0 = kt * 4 + kh * 2;
#pragma unroll
        for (int nt = 0; nt < 4; ++nt) {
            int col = nt * 16 + m;
            if (kt < 2) { bf[kt][nt].x = W1l[k0 * HID + col];
                          bf[kt][nt].y = W1l[(k0 + 1) * HID + col]; }
            else        { bf[kt][nt].x = W1r[(k0 - 8) * HID + col];
                          bf[kt][nt].y = W1r[(k0 - 7) * HID + col]; }
        }
    }
#pragma unroll
    for (int nt = 0; nt < 4; ++nt) bias[nt] = b1[nt * 16 + m];

    const int ntiles = (N + 15) >> 4;
    const int wpb = blockDim.x >> 5;
    for (int tile = blockIdx.x * wpb + wave; tile < ntiles; tile += gridDim.x * wpb) {
        const int tU = __builtin_amdgcn_readfirstlane(tile);   // scalarize tile index
        int row  = tU * 16 + m;
        int rowc = row < N ? row : N - 1;                      // branchless clamp
        float ic = __builtin_amdgcn_rcpf(fmaxf(cnt[rowc], 1.0f));

        v8f acc[4] = {};
#pragma unroll
        for (int kt = 0; kt < 4; ++kt) {
            int k0 = kt * 4 + kh * 2;
            v2f a;
            if (kt < 2) {                              // aggregated-neighbor half
                v2f t = *reinterpret_cast<const v2f*>(aggr1 + (long long)rowc * IN_CH + k0);
                a.x = t.x * ic; a.y = t.y * ic;
            } else {                                   // root-feature half
                a = *reinterpret_cast<const v2f*>(x + (long long)rowc * IN_CH + (k0 - 8));
            }
#pragma unroll
            for (int nt = 0; nt < 4; ++nt)
                acc[nt] = __builtin_amdgcn_wmma_f32_16x16x4_f32(
                    false, a, false, bf[kt][nt], (short)0, acc[nt], false, false);
        }
        // Epilogue: uniform branch between full tile (always for N%16==0) and tail
        if (tU * 16 + 16 <= N) {
#pragma unroll
            for (int nt = 0; nt < 4; ++nt) {
                float* hp = h + ((long long)tU * 16 + kh * 8) * HID + nt * 16 + m;
#pragma unroll
                for (int r = 0; r < 8; ++r)
                    hp[r * HID] = fmaxf(acc[nt][r] + bias[nt], 0.0f);
            }
        } else {
#pragma unroll
            for (int nt = 0; nt < 4; ++nt)
#pragma unroll
                for (int r = 0; r < 8; ++r) {
                    int orow = tU * 16 + r + kh * 8;
                    if (orow < N)
                        h[(long long)orow * HID + nt * 16 + m] =
                            fmaxf(acc[nt][r] + bias[nt], 0.0f);
                }
        }
    }
}

// ---------------------------------------------------------------------------
// TDM: stage a [64x32] f32 row-major matrix into LDS with a 32-dword pad after
// every 32-dword row -> rows land on a 64-float stride (interleaved layout).
// ---------------------------------------------------------------------------
#if __has_builtin(__builtin_amdgcn_tensor_load_to_lds)
#define HAVE_TDM 1
__device__ __forceinline__ void tdm_load_w(const float* gsrc, unsigned lds_byte_off) {
    unsigned long long ga = (unsigned long long)(uintptr_t)gsrc;
    v4u g0;
    g0.x = 1u;                                                  // count=1, user desc
    g0.y = lds_byte_off;                                        // lds_addr
    g0.z = (unsigned)(ga & 0xFFFFFFFFu);                        // global_addr[31:0]
    g0.w = (unsigned)((ga >> 32) & 0x01FFFFFFu) | (2u << 30);   // ga[56:32] | type=2
    v8i g1;
    g1[0] = (int)((2u << 16)            // data_size = 4B
                | (1u << 20)            // pad_enable
                | (4u << 22)            // pad_interval: 32 dwords
                | (31u << 25));         // pad_amount:   32 dwords
    g1[1] = (int)(32u << 16);           // tensor_dim0 = 32
    g1[2] = (int)(64u << 16);           // tensor_dim1 = 64
    g1[3] = (int)(32u << 16);           // tile_dim0 = 32
    g1[4] = 64;                         // tile_dim1 = 64
    g1[5] = 32;                         // tensor_dim0_stride = 32
    g1[6] = 0; g1[7] = 0;
    v4i gz = {0, 0, 0, 0};
#if defined(__clang_major__) && __clang_major__ >= 23
    v8i gz8 = {0, 0, 0, 0, 0, 0, 0, 0};
    __builtin_amdgcn_tensor_load_to_lds(g0, g1, gz, gz, gz8, 0);
#else
    __builtin_amdgcn_tensor_load_to_lds(g0, g1, gz, gz, 0);
#endif
}
#else
#define HAVE_TDM 0
#endif

// ---------------------------------------------------------------------------
// GEMM2: hlr = h (Nx64) @ [W2l | W2r] (64x64), weights TDM-staged in LDS.
// K=64 -> 16 WMMA f32 16x16x4 steps per col-tile.
// ---------------------------------------------------------------------------
__global__ void k_gemm2(const float* __restrict__ h,
                        const float* __restrict__ W2l,   // [64,32]
                        const float* __restrict__ W2r,   // [64,32]
                        float* __restrict__ hlr, int N) {
    __shared__ float sW[64 * 64];
#if HAVE_TDM
    if (threadIdx.x < 32) {          // wave 0 issues both tensor DMAs
        unsigned base = (unsigned)(unsigned long long)(uintptr_t)(void*)sW;
        tdm_load_w(W2l, base);           // -> sW[k*64 + 0..31]
        tdm_load_w(W2r, base + 128);     // -> sW[k*64 + 32..63]
        __builtin_amdgcn_s_wait_tensorcnt(0);
    }
#else
    for (int i = threadIdx.x; i < 64 * 64; i += blockDim.x) {
        int k = i >> 6, c = i & 63;
        sW[i] = (c < EMB) ? W2l[k * EMB + c] : W2r[k * EMB + (c - EMB)];
    }
#endif
    __syncthreads();

    const int lane = threadIdx.x & 31;
    const int wave = threadIdx.x >> 5;
    const int m  = lane & 15;
    const int kh = lane >> 4;

    const int ntiles = (N + 15) >> 4;
    const int wpb = blockDim.x >> 5;
    for (int tile = blockIdx.x * wpb + wave; tile < ntiles; tile += gridDim.x * wpb) {
        const int tU = __builtin_amdgcn_readfirstlane(tile);
        int row  = tU * 16 + m;
        int rowc = row < N ? row : N - 1;
        const float* hrow = h + (long long)rowc * HID;

        v8f acc[4] = {};
#pragma unroll
        for (int kt = 0; kt < 16; ++kt) {
            int k0 = kt * 4 + kh * 2;
            v2f a = *reinterpret_cast<const v2f*>(hrow + k0);
#pragma unroll
            for (int nt = 0; nt < 4; ++nt) {
                int col = nt * 16 + m;
                v2f b;
                b.x = sW[k0 * 64 + col];
                b.y = sW[(k0 + 1) * 64 + col];
                acc[nt] = __builtin_amdgcn_wmma_f32_16x16x4_f32(
                    false, a, false, b, (short)0, acc[nt], false, false);
            }
        }
        if (tU * 16 + 16 <= N) {
#pragma unroll
            for (int nt = 0; nt < 4; ++nt) {
                float* op = hlr + ((long long)tU * 16 + kh * 8) * 64 + nt * 16 + m;
#pragma unroll
                for (int r = 0; r < 8; ++r)
                    op[r * 64] = acc[nt][r];
            }
        } else {
#pragma unroll
            for (int nt = 0; nt < 4; ++nt)
#pragma unroll
                for (int r = 0; r < 8; ++r) {
                    int orow = tU * 16 + r + kh * 8;
                    if (orow < N)
                        hlr[(long long)orow * 64 + nt * 16 + m] = acc[nt][r];
                }
        }
    }
}

// ---------------------------------------------------------------------------
// Layer-2 edge scatter: aggr2[dst] += hlr[src, 0:32]
// ---------------------------------------------------------------------------
__global__ void k_edge2(const float* __restrict__ hlr,
                        const int* __restrict__ eidx, int E,
                        float* __restrict__ aggr2) {
    long long tid = (long long)blockIdx.x * blockDim.x + threadIdx.x;
    if (tid >= (long long)E * 8) return;
    int e = (int)(tid >> 3);
    int g = (int)(tid & 7);
    int src = eidx[e];
    int dst = eidx[E + e];
    float4 v = *reinterpret_cast<const float4*>(hlr + (long long)src * 64 + g * 4);
    float* a = aggr2 + (long long)dst * EMB + g * 4;
    atomicAdd(a + 0, v.x);
    atomicAdd(a + 1, v.y);
    atomicAdd(a + 2, v.z);
    atomicAdd(a + 3, v.w);
}

// ---------------------------------------------------------------------------
// Node epilogue + head:
//   z = relu(aggr2/cnt + b2 + hlr[:,32:64]);  out = sigmoid(relu(z@Wh1+bh1)@Wh2+bh2)
// ---------------------------------------------------------------------------
__global__ void k_node2_head(const float* __restrict__ hlr,
                             const float* __restrict__ aggr2,
                             const float* __restrict__ cnt,
                             const float* __restrict__ b2,    // [32]
                             const float* __restrict__ Wh1,   // [32,16]
                             const float* __restrict__ bh1,   // [16]
                             const float* __restrict__ Wh2,   // [16,1]
                             const float* __restrict__ bh2,   // [1]
                             float* __restrict__ out, int N) {
    int i = blockIdx.x * blockDim.x + threadIdx.x;
    if (i >= N) return;
    float ic = __builtin_amdgcn_rcpf(fmaxf(cnt[i], 1.0f));
    const float* hr = hlr + (long long)i * 64 + EMB;
    const float* ag = aggr2 + (long long)i * EMB;

    float t[16];
#pragma unroll
    for (int j = 0; j < 16; ++j) t[j] = bh1[j];
#pragma unroll
    for (int c = 0; c < EMB; ++c) {
        float z = fmaxf(ag[c] * ic + b2[c] + hr[c], 0.0f);
#pragma unroll
        for (int j = 0; j < 16; ++j) t[j] = fmaf(z, Wh1[c * 16 + j], t[j]);
    }
    float logit = bh2[0];
#pragma unroll
    for (int j = 0; j < 16; ++j) logit = fmaf(fmaxf(t[j], 0.0f), Wh2[j], logit);
    out[i] = 1.0f / (1.0f + __expf(-logit));
}

// ---------------------------------------------------------------------------
extern "C" void kernel_launch(void* const* d_in, const int* in_sizes, int n_in,
                              void* d_out, int out_size, void* d_ws, size_t ws_size,
                              hipStream_t stream) {
    const float* x   = (const float*)d_in[0];
    const int*   eix = (const int*)  d_in[1];
    const float* W1l = (const float*)d_in[2];
    const float* b1  = (const float*)d_in[3];
    const float* W1r = (const float*)d_in[4];
    const float* W2l = (const float*)d_in[5];
    const float* b2  = (const float*)d_in[6];
    const float* W2r = (const float*)d_in[7];
    const float* Wh1 = (const float*)d_in[8];
    const float* bh1 = (const float*)d_in[9];
    const float* Wh2 = (const float*)d_in[10];
    const float* bh2 = (const float*)d_in[11];
    float* out = (float*)d_out;

    const int N = in_sizes[0] / IN_CH;
    const int E = in_sizes[1] / 2;

    // Workspace (floats): aggr1[N*8] | cnt[N] | aggr2[N*32] | h[N*64] | hlr[N*64]
    float* ws    = (float*)d_ws;
    float* aggr1 = ws;
    float* cnt   = aggr1 + (size_t)N * IN_CH;
    float* aggr2 = cnt + N;
    float* h     = aggr2 + (size_t)N * EMB;
    float* hlr   = h + (size_t)N * HID;

    const int B = 256;

    long long nz = (long long)N * (IN_CH + 1 + EMB);
    k_zero<<<(int)((nz + B - 1) / B), B, 0, stream>>>(ws, nz);

    long long t1 = (long long)E * 2;
    k_edge1<<<(int)((t1 + B - 1) / B), B, 0, stream>>>(x, eix, E, aggr1, cnt);

    int ntiles = (N + 15) / 16;
    int wpb = B / 32;
    k_gemm1<<<(ntiles + wpb - 1) / wpb, B, 0, stream>>>(x, aggr1, cnt, W1l, b1, W1r, h, N);

    k_gemm2<<<(ntiles + wpb - 1) / wpb, B, 0, stream>>>(h, W2l, W2r, hlr, N);

    long long t2 = (long long)E * 8;
    k_edge2<<<(int)((t2 + B - 1) / B), B, 0, stream>>>(hlr, eix, E, aggr2);

    k_node2_head<<<(N + B - 1) / B, B, 0, stream>>>(hlr, aggr2, cnt, b2, Wh1, bh1, Wh2, bh2, out, N);
}